// K2_Net_2181843386886
// MI455X (gfx1250) — compile-verified
//
#include <hip/hip_runtime.h>

// CDNA5 / gfx1250 (MI455X): wave32, V_WMMA_F32_16X16X4_F32, async global->LDS.
// out_n = W1 (64x64) * X_n (64x64) * W2^T (64x64), n = 0..8191.
// Memory-bound: 256 MB @ 23.3 TB/s ~= 11 us floor. fp32 WMMA keeps fidelity.
// Weights are sample-invariant -> fragments live in registers (loaded from L2
// once); LDS holds only double-buffered X and the intermediate Y (48 KB).

typedef float v2f __attribute__((ext_vector_type(2)));
typedef float v8f __attribute__((ext_vector_type(8)));

#define DIM     64
#define THREADS 256   // 8 waves

// LDS float offsets: X0 @ 0, X1 @ 4096, Y @ 8192  (12288 floats = 48 KB)
#define XBUF0 0
#define XBUF1 4096
#define YBUF  8192

__device__ __forceinline__ void async_load_b128(unsigned lds_byte_off,
                                                const void* gaddr) {
    // ASYNCcnt-tracked DMA: copies 16 bytes per lane from the global address in vaddr to the LDS address in vdst.
    asm volatile("global_load_async_to_lds_b128 %0, %1, off"
                 :
                 : "v"(lds_byte_off),
                   "v"((unsigned long long)(uintptr_t)gaddr)
                 : "memory");
}

__device__ __forceinline__ void wait_async0() {
    asm volatile("s_wait_asynccnt 0x0" ::: "memory");
}

__global__ __launch_bounds__(THREADS) void kron_orth_kernel(
    const float* __restrict__ data,
    const float* __restrict__ W1,
    const float* __restrict__ W2,
    float* __restrict__ out,
    int nsamples)
{
    __shared__ float lds[12288];   // single LDS object -> starts at offset 0

    const int tid  = threadIdx.x;
    const int w    = tid >> 5;     // wave 0..7
    const int lane = tid & 31;
    const int m    = lane & 15;    // row (A) / col (B,C,D) index within tile
    const int g    = lane >> 4;    // half-wave group

    // Tile ownership (16 tiles of 16x16 per GEMM, 2 per wave):
    const int jt  = w >> 1;        // GEMM1: Y row tile (X read once block-wide)
    const int ktp = w & 1;         //        produces kt tiles {2ktp, 2ktp+1}
    const int kt2 = w >> 1;        // GEMM2: out col tile (Y read once block-wide)
    const int itp = w & 1;         //        produces it tiles {2itp, 2itp+1}

    // ---- Weight fragments -> registers (global/L2 reads, once per kernel) ----
    // B layout (KxN): vgpr v holds row K = v + 2g, col = m.
    // GEMM1 B = W2^T: b[v] = W2[kt*16+m, lt*4 + 2g + v]  (contiguous pair)
    v2f w2f[2][16];
#pragma unroll
    for (int q = 0; q < 2; ++q)
#pragma unroll
        for (int lt = 0; lt < 16; ++lt)
            w2f[q][lt] = *(const v2f*)&W2[((2 * ktp + q) * 16 + m) * DIM
                                          + lt * 4 + 2 * g];

    // A layout (16x4): lane m holds A[m, 2g + v] for K-block jt2.
    // GEMM2 A = W1: a[v] = W1[it*16+m, jt2*4 + 2g + v]  (contiguous pair)
    v2f w1f[2][16];
#pragma unroll
    for (int q = 0; q < 2; ++q)
#pragma unroll
        for (int jt2 = 0; jt2 < 16; ++jt2)
            w1f[q][jt2] = *(const v2f*)&W1[((2 * itp + q) * 16 + m) * DIM
                                           + jt2 * 4 + 2 * g];

    const int n0     = blockIdx.x;
    const int stride = gridDim.x;

    // ---- Pipeline prologue: async-stage first sample into X buffer 0 ----
    if (n0 < nsamples) {
        const char* Xg = (const char*)(data + (size_t)n0 * (DIM * DIM));
#pragma unroll
        for (int t = 0; t < 4; ++t) {
            int idx = t * THREADS + tid;           // 1024 x 16B = 16 KB
            async_load_b128(XBUF0 * 4 + idx * 16, Xg + idx * 16);
        }
    }

    int cur = 0;
    for (int n = n0; n < nsamples; n += stride) {
        wait_async0();        // current X buffer landed (this wave's loads)
        __syncthreads();      // all waves landed; prev X buf + Y free for reuse

        // ---- Async-stage next sample into the other X buffer ----
        {
            int nn = n + stride;
            if (nn < nsamples) {
                const char* Xg = (const char*)(data + (size_t)nn * (DIM * DIM));
                unsigned xb = (cur ? XBUF0 : XBUF1) * 4;
#pragma unroll
                for (int t = 0; t < 4; ++t) {
                    int idx = t * THREADS + tid;
                    async_load_b128(xb + idx * 16, Xg + idx * 16);
                }
            }
        }

        const int xbase = cur ? XBUF1 : XBUF0;

        // ================= GEMM1: Y = X * W2^T =================
        // Wave computes Y rows jt*16..+15, cols (2ktp)*16 .. (2ktp+1)*16+15.
        {
            v8f c0 = {}, c1 = {};
#pragma unroll
            for (int lt = 0; lt < 16; ++lt) {
                // A frag: X[jt*16+m, lt*4 + 2g + {0,1}]
                v2f a = *(const v2f*)&lds[xbase + (jt * 16 + m) * DIM
                                          + lt * 4 + 2 * g];
                c0 = __builtin_amdgcn_wmma_f32_16x16x4_f32(
                         false, a, false, w2f[0][lt], (short)0, c0, false, false);
                c1 = __builtin_amdgcn_wmma_f32_16x16x4_f32(
                         false, a, false, w2f[1][lt], (short)0, c1, false, false);
            }
            // C/D layout: row = jt*16 + v + 8g, col = kt*16 + m
#pragma unroll
            for (int v = 0; v < 8; ++v) {
                int row = jt * 16 + v + 8 * g;
                lds[YBUF + row * DIM + (2 * ktp    ) * 16 + m] = c0[v];
                lds[YBUF + row * DIM + (2 * ktp + 1) * 16 + m] = c1[v];
            }
        }
        __syncthreads();

        // ================= GEMM2: out = W1 * Y =================
        // Wave computes out rows {2itp,2itp+1}*16 tiles, cols kt2*16..+15.
        {
            v8f c0 = {}, c1 = {};
#pragma unroll
            for (int jt2 = 0; jt2 < 16; ++jt2) {
                // B frag: Y[jt2*4 + 2g + v, kt2*16 + m] (two rows -> two b32)
                v2f b;
                int r0 = YBUF + (jt2 * 4 + 2 * g) * DIM + kt2 * 16 + m;
                b.x = lds[r0];
                b.y = lds[r0 + DIM];
                c0 = __builtin_amdgcn_wmma_f32_16x16x4_f32(
                         false, w1f[0][jt2], false, b, (short)0, c0, false, false);
                c1 = __builtin_amdgcn_wmma_f32_16x16x4_f32(
                         false, w1f[1][jt2], false, b, (short)0, c1, false, false);
            }
            float* og = out + (size_t)n * (DIM * DIM);
#pragma unroll
            for (int v = 0; v < 8; ++v) {
                og[((2 * itp    ) * 16 + v + 8 * g) * DIM + kt2 * 16 + m] = c0[v];
                og[((2 * itp + 1) * 16 + v + 8 * g) * DIM + kt2 * 16 + m] = c1[v];
            }
        }

        cur ^= 1;
        // loop-top barrier covers Y WAR and X-buffer WAR hazards
    }
}

extern "C" void kernel_launch(void* const* d_in, const int* in_sizes, int n_in,
                              void* d_out, int out_size, void* d_ws, size_t ws_size,
                              hipStream_t stream) {
    (void)n_in; (void)out_size; (void)d_ws; (void)ws_size;
    const float* data = (const float*)d_in[0];
    const float* W1   = (const float*)d_in[1];
    const float* W2   = (const float*)d_in[2];
    float* out        = (float*)d_out;

    int nsamples = in_sizes[0] / (DIM * DIM);   // 8192
    int blocks = 512;                           // ~2 blocks/WGP, 16 samples each
    if (blocks > nsamples) blocks = nsamples;

    hipLaunchKernelGGL(kron_orth_kernel, dim3(blocks), dim3(THREADS), 0, stream,
                       data, W1, W2, out, nsamples);
}